// OptNet_2525440770614
// MI455X (gfx1250) — compile-verified
//
#include <hip/hip_runtime.h>
#include <math.h>

// OptNet QP head on MI455X (gfx1250).
// The reference's fc1 GEMM is dead code -> skipped entirely. The live work is
// a 10x10 QP (single inequality, exact KKT) + log-softmax: launch-latency
// bound, so one wave32 does everything. The one matmul (tril(L) @ tril(L)^T)
// is padded to 16x16 and executed on the matrix pipe with
// V_WMMA_F32_16X16X4_F32 (4 K-chunks of 4).

typedef __attribute__((ext_vector_type(2))) float v2f;
typedef __attribute__((ext_vector_type(8))) float v8f;

#define NC   10
#define QEPS 1e-4f

__launch_bounds__(32)
__global__ void optnet_qp_kernel(const float* __restrict__ Lmat,
                                 const float* __restrict__ p,
                                 const float* __restrict__ G,
                                 const float* __restrict__ z0,
                                 const float* __restrict__ s0,
                                 float* __restrict__ out) {
  __shared__ float sLm[16][16];  // tril(L), zero-padded to 16x16
  __shared__ float sQ[16][16];   // Q = Lm*Lm^T + eps*I

  const int lane = threadIdx.x;  // 0..31, single full wave (EXEC all ones)

  // ---- Stage padded tril(L) into LDS (8 elements per lane) ----
  for (int i = 0; i < 8; ++i) {
    int idx = lane * 8 + i;
    int r = idx >> 4, c = idx & 15;
    float v = 0.0f;
    if (r < NC && c < NC && c <= r) v = Lmat[r * NC + c];
    sLm[r][c] = v;
  }
  __syncthreads();

  // ---- Q0 = Lm * Lm^T on the matrix pipe ----
  // A-fragment (16x4 f32): lane m=lane&15; VGPR0 = K=kb, VGPR1 = K=kb+1,
  // where kb = 4*kc + (lane<16 ? 0 : 2).
  // B = Lm^T, and the B-fragment layout is the lane-transpose of A's, so the
  // per-lane B values equal the per-lane A values: feed the same v2f twice.
  const int m     = lane & 15;
  const int khalf = (lane >> 4) * 2;
  v8f acc = {};
  for (int kc = 0; kc < 4; ++kc) {
    const int kb = kc * 4 + khalf;
    v2f a;
    a.x = sLm[m][kb];
    a.y = sLm[m][kb + 1];
    // 8 args: (neg_a, A, neg_b, B, c_mod, C, reuse_a, reuse_b)
    acc = __builtin_amdgcn_wmma_f32_16x16x4_f32(false, a, false, a,
                                                (short)0, acc, false, false);
  }

  // ---- Scatter C layout back to LDS, adding eps*I ----
  {
    const int mbase = (lane >> 4) * 8;  // lanes 16-31 hold M=8..15
    const int n     = lane & 15;
    for (int r = 0; r < 8; ++r) {
      const int M = mbase + r;
      float q = acc[r];
      if (M == n) q += QEPS;
      sQ[M][n] = q;
    }
  }
  __syncthreads();

  // ---- Tiny exact-KKT QP + log-softmax: scalar work on lane 0 ----
  if (lane == 0) {
    float A[NC][NC];
    for (int i = 0; i < NC; ++i)
      for (int j = 0; j < NC; ++j) A[i][j] = sQ[i][j];

    // Cholesky: A -> C (lower), Q = C C^T. SPD by construction (eps*I).
    for (int k = 0; k < NC; ++k) {
      const float d = sqrtf(A[k][k]);
      A[k][k] = d;
      const float inv = 1.0f / d;
      for (int i = k + 1; i < NC; ++i) A[i][k] *= inv;
      for (int j = k + 1; j < NC; ++j)
        for (int i = j; i < NC; ++i) A[i][j] -= A[i][k] * A[j][k];
    }

    float t[NC], zu[NC], y[NC];

    // z_unc = Q^{-1}(-p): forward then backward substitution
    for (int i = 0; i < NC; ++i) {
      float s = -p[i];
      for (int j = 0; j < i; ++j) s -= A[i][j] * t[j];
      t[i] = s / A[i][i];
    }
    for (int i = NC - 1; i >= 0; --i) {
      float s = t[i];
      for (int j = i + 1; j < NC; ++j) s -= A[j][i] * zu[j];
      zu[i] = s / A[i][i];
    }

    // y = Q^{-1} g
    for (int i = 0; i < NC; ++i) {
      float s = G[i];
      for (int j = 0; j < i; ++j) s -= A[i][j] * t[j];
      t[i] = s / A[i][i];
    }
    for (int i = NC - 1; i >= 0; --i) {
      float s = t[i];
      for (int j = i + 1; j < NC; ++j) s -= A[j][i] * y[j];
      y[i] = s / A[i][i];
    }

    // h = G@z0 + s0 ; lam = max(g@z_unc - h, 0) / (g@y) ; z = z_unc - lam*y
    float h = s0[0];
    for (int i = 0; i < NC; ++i) h += G[i] * z0[i];
    float viol = -h, denom = 0.0f;
    for (int i = 0; i < NC; ++i) {
      viol  += G[i] * zu[i];
      denom += G[i] * y[i];
    }
    const float lam = fmaxf(viol, 0.0f) / denom;

    // log-softmax
    float z[NC], mx = -INFINITY;
    for (int i = 0; i < NC; ++i) {
      z[i] = zu[i] - lam * y[i];
      mx = fmaxf(mx, z[i]);
    }
    float se = 0.0f;
    for (int i = 0; i < NC; ++i) se += expf(z[i] - mx);
    const float lse = logf(se);
    for (int i = 0; i < NC; ++i) out[i] = z[i] - mx - lse;
  }
}

extern "C" void kernel_launch(void* const* d_in, const int* in_sizes, int n_in,
                              void* d_out, int out_size, void* d_ws, size_t ws_size,
                              hipStream_t stream) {
  (void)in_sizes; (void)n_in; (void)out_size; (void)d_ws; (void)ws_size;
  // setup_inputs order: x, W1, b1, L, p, G, z0, s0
  const float* Lmat = (const float*)d_in[3];
  const float* p    = (const float*)d_in[4];
  const float* G    = (const float*)d_in[5];
  const float* z0   = (const float*)d_in[6];
  const float* s0   = (const float*)d_in[7];
  optnet_qp_kernel<<<1, 32, 0, stream>>>(Lmat, p, G, z0, s0, (float*)d_out);
}